// checkerboard_ZZ_13924283973961
// MI455X (gfx1250) — compile-verified
//
#include <hip/hip_runtime.h>

typedef __attribute__((ext_vector_type(2))) float v2f;
typedef __attribute__((ext_vector_type(8))) float v8f;

#define NQ      24
#define LAYERS  4
#define S_ELEMS (1u << NQ)   // 2^24 amplitudes
#define TILE    4096         // 16x16x16 cube per block
#define TPB     128          // 4 waves (wave32)

// LDS tile is a 16x16x16 cube: element t = a2*256 + a1*16 + a0.
// Axis g has stride 16^g; 'o' enumerates the 256 combined other-axis values.
__device__ __forceinline__ int tidx(int g, int c, int o) {
  int ol = o & 15, oh = o >> 4;
  if (g == 0) return (oh << 8) | (ol << 4) | c;   // o = (a2,a1)
  if (g == 1) return (oh << 8) | (c << 4) | ol;   // o = (a2,a0)
  return (c << 8) | (oh << 4) | ol;               // o = (a1,a0)
}

// ---------------- deterministic norm reduction ----------------
__global__ void k_norm_partial(const float* __restrict__ re,
                               const float* __restrict__ im,
                               float* __restrict__ partial) {
  __shared__ float red[256];
  int base = blockIdx.x * 4096;
  float acc = 0.f;
  for (int i = threadIdx.x; i < 4096; i += 256) {
    float r = re[base + i], m = im[base + i];
    acc = fmaf(r, r, fmaf(m, m, acc));
  }
  red[threadIdx.x] = acc;
  __syncthreads();
  for (int s = 128; s > 0; s >>= 1) {
    if (threadIdx.x < s) red[threadIdx.x] += red[threadIdx.x + s];
    __syncthreads();
  }
  if (threadIdx.x == 0) partial[blockIdx.x] = red[0];
}

__global__ void k_norm_final(const float* __restrict__ partial, float* __restrict__ norm) {
  __shared__ float red[256];
  float acc = 0.f;
  for (int i = threadIdx.x; i < 4096; i += 256) acc += partial[i];
  red[threadIdx.x] = acc;
  __syncthreads();
  for (int s = 128; s > 0; s >>= 1) {
    if (threadIdx.x < s) red[threadIdx.x] += red[threadIdx.x + s];
    __syncthreads();
  }
  if (threadIdx.x == 0) norm[0] = red[0];
}

// normalize + convert planar -> interleaved complex in d_out (our working buffer)
__global__ void k_scale_copy(const float* __restrict__ re, const float* __restrict__ im,
                             const float* __restrict__ norm, float2* __restrict__ out) {
  float s = rsqrtf(norm[0]);
  unsigned j = blockIdx.x * 256u + threadIdx.x;  // one float4 = 2 complex
  float4 v;
  v.x = re[2 * j] * s;     v.y = im[2 * j] * s;
  v.z = re[2 * j + 1] * s; v.w = im[2 * j + 1] * s;
  ((float4*)out)[j] = v;
}

// ---------------- closed-form 2x2 expm of (H - H^dagger) ----------------
// A = i*M, M Hermitian = t*I + x*sx + y*sy + z*sz ; exp(iM) = e^{it}(cos r + i sin r n.sigma)
__global__ void k_expm(const float* __restrict__ Hre, const float* __restrict__ Him,
                       float* __restrict__ U) {
  int t = blockIdx.x * blockDim.x + threadIdx.x;
  if (t >= LAYERS * NQ) return;
  const float* hr = Hre + t * 4;
  const float* hi = Him + t * 4;
  float x  = hi[1] + hi[2];   // Re(M01)
  float y  = hr[1] - hr[2];   // -Im(M01)
  float tt = hi[0] + hi[3];   // trace/2 * 2
  float z  = hi[0] - hi[3];
  float r  = sqrtf(x * x + y * y + z * z);
  float cr = cosf(r);
  float s  = (r > 1e-20f) ? (sinf(r) / r) : 1.0f;
  float ct = cosf(tt), st = sinf(tt);
  float u[8] = { cr,  z * s,    // u00
                 y * s,  x * s, // u01
                -y * s,  x * s, // u10
                 cr, -z * s };  // u11
  float* o = U + t * 8;
#pragma unroll
  for (int k = 0; k < 4; ++k) {
    o[2 * k]     = u[2 * k] * ct - u[2 * k + 1] * st;
    o[2 * k + 1] = u[2 * k] * st + u[2 * k + 1] * ct;
  }
}

// Kronecker product of 4 single-qubit gates -> 16x16 complex W per (layer, group)
// flat-index bit b corresponds to qubit 23-b; group g covers bits [4g, 4g+4)
__global__ void k_kron(const float* __restrict__ U, float* __restrict__ W) {
  int lg = blockIdx.x;           // l*6 + g
  int l = lg / 6, g = lg % 6;
  int rc = threadIdx.x;          // r*16 + c
  int r = rc >> 4, c = rc & 15;
  float wr = 1.f, wi = 0.f;
#pragma unroll
  for (int k = 0; k < 4; ++k) {
    int q = 23 - (4 * g + k);
    int rk = (r >> k) & 1, ck = (c >> k) & 1;
    const float* u = U + (l * NQ + q) * 8 + (rk * 2 + ck) * 2;
    float nr = wr * u[0] - wi * u[1];
    float ni = wr * u[1] + wi * u[0];
    wr = nr; wi = ni;
  }
  float* wb = W + lg * 512;      // 256 re, then 256 im
  wb[rc]       = wr;
  wb[256 + rc] = wi;
}

// ---------------- main in-place sweep: 3 fused 16x16 complex WMMA matmuls ----------------
__global__ __launch_bounds__(TPB)
void k_pass(float2* __restrict__ st, const float* __restrict__ W3,
            const float* __restrict__ th1, const float* __restrict__ th2, int highpass) {
  __shared__ float lre[TILE];
  __shared__ float lim[TILE];
  __shared__ float th[24];
  const int tid  = threadIdx.x;
  const int lane = tid & 31;
  const int wave = tid >> 5;
  const int half = lane >> 4;
  const int l16  = lane & 15;
  const unsigned blk = blockIdx.x;

  if (highpass && tid < 12) { th[tid] = th1[tid]; th[tid + 12] = th2[tid]; }

  if (!highpass) {
    const float4* g4 = (const float4*)(st + (size_t)blk * TILE);
    for (int i = tid; i < TILE / 2; i += TPB) {
      float4 v = g4[i];
      lre[2 * i] = v.x; lim[2 * i] = v.y;
      lre[2 * i + 1] = v.z; lim[2 * i + 1] = v.w;
    }
  } else {
    for (int t = tid; t < TILE; t += TPB) {
      float2 v = st[((size_t)t << 12) | blk];   // scattered, L2-resident
      lre[t] = v.x; lim[t] = v.y;
    }
  }
  __syncthreads();

  for (int g = 0; g < 3; ++g) {
    const float* Wr = W3 + g * 512;
    const float* Wi = Wr + 256;
    // A fragments (16x4 f32 per K-step): lane<16 -> K=4kk+{0,1}; lane>=16 -> K=4kk+{2,3}
    v2f ar[4], ai[4], an[4];
#pragma unroll
    for (int kk = 0; kk < 4; ++kk) {
      int c0 = kk * 4 + half * 2;
      v2f tr = { Wr[l16 * 16 + c0], Wr[l16 * 16 + c0 + 1] };
      v2f ti = { Wi[l16 * 16 + c0], Wi[l16 * 16 + c0 + 1] };
      ar[kk] = tr; ai[kk] = ti; an[kk] = -ti;
    }
#pragma unroll
    for (int jt = 0; jt < 4; ++jt) {
      int o = (wave + 4 * jt) * 16 + l16;   // this wave's 16-column tile
      v8f cr = {}; v8f ci = {};
#pragma unroll
      for (int kk = 0; kk < 4; ++kk) {
        int c0 = kk * 4 + half * 2;
        int i0 = tidx(g, c0, o), i1 = tidx(g, c0 + 1, o);
        v2f br = { lre[i0], lre[i1] };
        v2f bi = { lim[i0], lim[i1] };
        cr = __builtin_amdgcn_wmma_f32_16x16x4_f32(false, ar[kk], false, br, (short)0, cr, false, false);
        cr = __builtin_amdgcn_wmma_f32_16x16x4_f32(false, an[kk], false, bi, (short)0, cr, false, false);
        ci = __builtin_amdgcn_wmma_f32_16x16x4_f32(false, ar[kk], false, bi, (short)0, ci, false, false);
        ci = __builtin_amdgcn_wmma_f32_16x16x4_f32(false, ai[kk], false, br, (short)0, ci, false, false);
      }
      // D layout: element j -> row j + 8*half, col l16 ; same 256 slots we read -> in-place OK
#pragma unroll
      for (int j = 0; j < 8; ++j) {
        int di = tidx(g, j + 8 * half, o);
        lre[di] = cr[j];
        lim[di] = ci[j];
      }
    }
    __syncthreads();
  }

  if (!highpass) {
    float4* g4 = (float4*)(st + (size_t)blk * TILE);
    for (int i = tid; i < TILE / 2; i += TPB)
      g4[i] = make_float4(lre[2 * i], lim[2 * i], lre[2 * i + 1], lim[2 * i + 1]);
  } else {
    // fold ALL diagonal ZZ phases of this layer in for free
    for (int t = tid; t < TILE; t += TPB) {
      unsigned n = ((unsigned)t << 12) | blk;
      float re = lre[t], im = lim[t];
      float phi = 0.f;
#pragma unroll
      for (int i = 0; i < 12; ++i) {
        int ba = 23 - 2 * i;                       // qubit 2i
        int x1 = (int)((n >> ba) ^ (n >> (ba - 1))) & 1;  // vs qubit 2i+1
        phi += th[i] * (1.f - 2.f * (float)x1);
        int bc = 22 - 2 * i;                       // qubit 2i+1
        int bd = (i == 11) ? 23 : (21 - 2 * i);    // qubit (2i+2)%24
        int x2 = (int)((n >> bc) ^ (n >> bd)) & 1;
        phi += th[12 + i] * (1.f - 2.f * (float)x2);
      }
      float sp, cp;
      __sincosf(phi, &sp, &cp);
      st[n] = make_float2(re * cp - im * sp, re * sp + im * cp);
    }
  }
}

extern "C" void kernel_launch(void* const* d_in, const int* in_sizes, int n_in,
                              void* d_out, int out_size, void* d_ws, size_t ws_size,
                              hipStream_t stream) {
  (void)in_sizes; (void)n_in; (void)out_size; (void)ws_size;
  const float* state_re = (const float*)d_in[0];
  const float* state_im = (const float*)d_in[1];
  const float* H_re     = (const float*)d_in[2];
  const float* H_im     = (const float*)d_in[3];
  const float* theta1   = (const float*)d_in[4];
  const float* theta2   = (const float*)d_in[5];

  float* ws      = (float*)d_ws;
  float* W       = ws;             // 24 * 512 = 12288 floats
  float* U       = ws + 12288;     // 96 * 8   = 768
  float* partial = ws + 13056;     // 4096
  float* norm    = ws + 17152;     // 1
  float2* st     = (float2*)d_out; // state lives interleaved in d_out, in-place

  k_norm_partial<<<4096, 256, 0, stream>>>(state_re, state_im, partial);
  k_norm_final<<<1, 256, 0, stream>>>(partial, norm);
  k_scale_copy<<<(S_ELEMS / 2) / 256, 256, 0, stream>>>(state_re, state_im, norm, st);
  k_expm<<<1, 128, 0, stream>>>(H_re, H_im, U);
  k_kron<<<LAYERS * 6, 256, 0, stream>>>(U, W);

  for (int l = 0; l < LAYERS; ++l) {
    k_pass<<<4096, TPB, 0, stream>>>(st, W + (l * 6 + 0) * 512,
                                     theta1 + l * 12, theta2 + l * 12, 0);
    k_pass<<<4096, TPB, 0, stream>>>(st, W + (l * 6 + 3) * 512,
                                     theta1 + l * 12, theta2 + l * 12, 1);
  }
}